// GINClassifier_33346126086713
// MI455X (gfx1250) — compile-verified
//
#include <hip/hip_runtime.h>
#include <hip/hip_bf16.h>

typedef __attribute__((ext_vector_type(2))) float v2f;
typedef __attribute__((ext_vector_type(8))) float v8f;

#define N_NODES 200000
#define N_EDGES 3200000
#define F_IN    128
#define H_DIM   64
#define G_NUM   512
#define C_NUM   10
#define BN_EPS  1e-5f

// ---------------------------------------------------------------- zero fill
__global__ void zero_f32(float* __restrict__ p, long long n) {
    long long i = (long long)blockIdx.x * blockDim.x + threadIdx.x;
    long long stride = (long long)gridDim.x * blockDim.x;
    for (; i < n; i += stride) p[i] = 0.0f;
}

// ------------------------------------------------- edge scatter-add (segment_sum)
// one thread per (edge, feature); consecutive lanes share an edge -> coalesced
// gather of x[src] rows and row-contiguous float atomics into agg[dst].
__global__ void scatter_add_feat(const float* __restrict__ x,
                                 const int* __restrict__ ei,   // [2, E]
                                 float* __restrict__ agg,
                                 int E, int Fdim, int logF) {
    long long i = (long long)blockIdx.x * blockDim.x + threadIdx.x;
    long long total = (long long)E << logF;
    if (i >= total) return;
    int e = (int)(i >> logF);
    int f = (int)(i & (Fdim - 1));
    int s = ei[e];
    int d = ei[E + e];
    atomicAdd(&agg[(long long)d * Fdim + f], x[(long long)s * Fdim + f]);
}

// ------------------------------------- fused (A0[+A1]) @ W + b -> BN(eval) -> ReLU
// C = relu(((A @ W) + bias) * gamma/sqrt(1+eps) + beta),  W is [K x 64].
// 256 threads = 8 waves; wave w -> row-tile (w>>2), col-tile (w&3).
// Each wave computes one 16x16 f32 tile via V_WMMA_F32_16X16X4_F32.
// A 16x4 layout : lanes 0-15 hold K={0,1}, lanes 16-31 hold K={2,3} (ISA 7.12.2)
// B 4x16 layout : symmetric (lane = column, vgpr = K within half)
// C/D layout    : vgpr r, lanes 0-15 -> M=r ; lanes 16-31 -> M=r+8
template <int K, bool ADD>
__global__ __launch_bounds__(256) void gemm_bn_relu(
        const float* __restrict__ A0, const float* __restrict__ A1,
        const float* __restrict__ W,
        const float* __restrict__ bias, const float* __restrict__ gam,
        const float* __restrict__ bet, float* __restrict__ out, int nrows) {
    const int HO = 64;
    int wave = threadIdx.x >> 5;          // wave32
    int lane = threadIdx.x & 31;
    int lh   = lane >> 4;                 // half-wave: selects K pair
    int l    = lane & 15;                 // row (A) / col (B,C,D) within tile
    int rowTile = (blockIdx.x * 2 + (wave >> 2)) * 16;
    int col0    = (wave & 3) * 16;
    if (rowTile >= nrows) return;         // wave-granular guard (EXEC stays all-1)

    const float* a0 = A0 + (size_t)(rowTile + l) * K + 2 * lh;
    const float* a1 = ADD ? (A1 + (size_t)(rowTile + l) * K + 2 * lh) : nullptr;
    const float* wp = W + (size_t)(2 * lh) * HO + col0 + l;

    v8f acc = {};
#pragma unroll
    for (int k = 0; k < K; k += 4) {
        v2f av, bv;
        av.x = a0[k];
        av.y = a0[k + 1];
        if (ADD) { av.x += a1[k]; av.y += a1[k + 1]; }
        bv.x = wp[(size_t)k * HO];
        bv.y = wp[(size_t)k * HO + HO];
        acc = __builtin_amdgcn_wmma_f32_16x16x4_f32(
                false, av, false, bv, (short)0, acc, false, false);
    }

    int   ocol = col0 + l;
    float sc = gam[ocol] * rsqrtf(1.0f + BN_EPS);
    float bb = bias[ocol];
    float bt = bet[ocol];
    float* op = out + (size_t)(rowTile + 8 * lh) * HO + ocol;
#pragma unroll
    for (int r = 0; r < 8; ++r) {
        float v = (acc[r] + bb) * sc + bt;
        op[(size_t)r * HO] = v > 0.0f ? v : 0.0f;
    }
}

// ---------------------------------------------------------------- mean pool
__global__ void pool_scatter(const float* __restrict__ h2,
                             const int* __restrict__ batch,
                             float* __restrict__ sums, int n) {
    long long i = (long long)blockIdx.x * blockDim.x + threadIdx.x;
    if (i >= (long long)n * H_DIM) return;
    int node = (int)(i >> 6);
    int f    = (int)(i & 63);
    atomicAdd(&sums[(long long)batch[node] * H_DIM + f], h2[i]);
}

__global__ void pool_count(const int* __restrict__ batch, float* __restrict__ cnt, int n) {
    int i = blockIdx.x * blockDim.x + threadIdx.x;
    if (i < n) atomicAdd(&cnt[batch[i]], 1.0f);
}

__global__ void pool_div(float* __restrict__ sums, const float* __restrict__ cnt, int g) {
    int i = blockIdx.x * blockDim.x + threadIdx.x;
    if (i < g * H_DIM) sums[i] /= fmaxf(cnt[i >> 6], 1.0f);
}

// ------------------------------------------- 64->10 projection + log_softmax
__global__ void head_kernel(const float* __restrict__ z,
                            const float* __restrict__ wc2,
                            const float* __restrict__ bc2,
                            float* __restrict__ out, int g) {
    int gi = blockIdx.x * blockDim.x + threadIdx.x;
    if (gi >= g) return;
    float logits[C_NUM];
#pragma unroll
    for (int c = 0; c < C_NUM; ++c) logits[c] = bc2[c];
    const float* zr = z + (size_t)gi * H_DIM;
    for (int k = 0; k < H_DIM; ++k) {
        float zv = zr[k];
#pragma unroll
        for (int c = 0; c < C_NUM; ++c) logits[c] += zv * wc2[k * C_NUM + c];
    }
    float m = logits[0];
#pragma unroll
    for (int c = 1; c < C_NUM; ++c) m = fmaxf(m, logits[c]);
    float s = 0.0f;
#pragma unroll
    for (int c = 0; c < C_NUM; ++c) s += __expf(logits[c] - m);
    float lse = __logf(s);
#pragma unroll
    for (int c = 0; c < C_NUM; ++c) out[(size_t)gi * C_NUM + c] = logits[c] - m - lse;
}

// ---------------------------------------------------------------- launcher
extern "C" void kernel_launch(void* const* d_in, const int* in_sizes, int n_in,
                              void* d_out, int out_size, void* d_ws, size_t ws_size,
                              hipStream_t stream) {
    const float* x    = (const float*)d_in[0];
    const int*   ei   = (const int*)d_in[1];
    const int*   batch= (const int*)d_in[2];
    const float *w1a = (const float*)d_in[3],  *b1a = (const float*)d_in[4];
    const float *w1b = (const float*)d_in[5],  *b1b = (const float*)d_in[6];
    const float *w2a = (const float*)d_in[7],  *b2a = (const float*)d_in[8];
    const float *w2b = (const float*)d_in[9],  *b2b = (const float*)d_in[10];
    const float *wc1 = (const float*)d_in[11], *bc1 = (const float*)d_in[12];
    const float *wc2 = (const float*)d_in[13], *bc2 = (const float*)d_in[14];
    const float *g1a = (const float*)d_in[15], *be1a = (const float*)d_in[16];
    const float *g1b = (const float*)d_in[17], *be1b = (const float*)d_in[18];
    const float *g2a = (const float*)d_in[19], *be2a = (const float*)d_in[20];
    const float *g2b = (const float*)d_in[21], *be2b = (const float*)d_in[22];
    const float *gc  = (const float*)d_in[23], *bec  = (const float*)d_in[24];
    float* out = (float*)d_out;

    // workspace layout (floats); agg1 region [N*128] is reused after conv1.
    float* ws   = (float*)d_ws;
    float* agg1 = ws;                                    // N*128
    float* h1   = agg1 + (size_t)N_NODES * F_IN;         // N*64
    float* h    = h1   + (size_t)N_NODES * H_DIM;        // N*64
    float* agg2 = agg1;                                  // reuse: N*64
    float* h2   = agg1 + (size_t)N_NODES * H_DIM;        // reuse: N*64
    float* t    = h1;                                    // reuse: N*64
    float* sums = h    + (size_t)N_NODES * H_DIM;        // G*64
    float* cnt  = sums + (size_t)G_NUM * H_DIM;          // G
    float* z    = cnt  + G_NUM;                          // G*64

    // ---- conv1: agg = segment_sum(x[src], dst);  h = MLP(x + agg)
    zero_f32<<<2048, 256, 0, stream>>>(agg1, (long long)N_NODES * F_IN);
    {
        long long tot = (long long)N_EDGES * F_IN;
        int blocks = (int)((tot + 255) / 256);
        scatter_add_feat<<<blocks, 256, 0, stream>>>(x, ei, agg1, N_EDGES, F_IN, 7);
    }
    gemm_bn_relu<F_IN, true ><<<N_NODES / 32, 256, 0, stream>>>(x,  agg1, w1a, b1a, g1a, be1a, h1, N_NODES);
    gemm_bn_relu<H_DIM, false><<<N_NODES / 32, 256, 0, stream>>>(h1, nullptr, w1b, b1b, g1b, be1b, h,  N_NODES);

    // ---- conv2
    zero_f32<<<2048, 256, 0, stream>>>(agg2, (long long)N_NODES * H_DIM);
    {
        long long tot = (long long)N_EDGES * H_DIM;
        int blocks = (int)((tot + 255) / 256);
        scatter_add_feat<<<blocks, 256, 0, stream>>>(h, ei, agg2, N_EDGES, H_DIM, 6);
    }
    gemm_bn_relu<H_DIM, true ><<<N_NODES / 32, 256, 0, stream>>>(h, agg2,   w2a, b2a, g2a, be2a, t,  N_NODES);
    gemm_bn_relu<H_DIM, false><<<N_NODES / 32, 256, 0, stream>>>(t, nullptr, w2b, b2b, g2b, be2b, h2, N_NODES);

    // ---- global mean pool
    zero_f32<<<64, 256, 0, stream>>>(sums, (long long)(G_NUM * H_DIM + G_NUM));
    {
        long long tot = (long long)N_NODES * H_DIM;
        int blocks = (int)((tot + 255) / 256);
        pool_scatter<<<blocks, 256, 0, stream>>>(h2, batch, sums, N_NODES);
    }
    pool_count<<<(N_NODES + 255) / 256, 256, 0, stream>>>(batch, cnt, N_NODES);
    pool_div<<<(G_NUM * H_DIM + 255) / 256, 256, 0, stream>>>(sums, cnt, G_NUM);

    // ---- classifier head
    gemm_bn_relu<H_DIM, false><<<G_NUM / 32, 256, 0, stream>>>(sums, nullptr, wc1, bc1, gc, bec, z, G_NUM);
    head_kernel<<<(G_NUM + 255) / 256, 256, 0, stream>>>(z, wc2, bc2, out, G_NUM);
}